// GraphEncoder_18528488914973
// MI455X (gfx1250) — compile-verified
//
#include <hip/hip_runtime.h>
#include <hip/hip_bf16.h>

typedef float v2f  __attribute__((ext_vector_type(2)));
typedef float v8f  __attribute__((ext_vector_type(8)));

#define TILE_COLS 64        // column stripe per block (4 WMMA tiles wide)
#define MAX_K     256

// ---------------------------------------------------------------------------
// fill with zeros
// ---------------------------------------------------------------------------
__global__ void fill_zero(float* __restrict__ p, long long n) {
    long long i = (long long)blockIdx.x * blockDim.x + threadIdx.x;
    if (i < n) p[i] = 0.0f;
}

// ---------------------------------------------------------------------------
// edge scatter: aggr[dst] += ew * x[src]   (one wave32 per edge, float4 lanes)
// ---------------------------------------------------------------------------
__global__ void scatter_edges(const float* __restrict__ x,
                              const long long* __restrict__ ei,   // [2, E]
                              const float* __restrict__ ew,
                              float* __restrict__ aggr,
                              int E, int C) {
    int wid  = (int)(((long long)blockIdx.x * blockDim.x + threadIdx.x) >> 5);
    int lane = threadIdx.x & 31;
    if (wid >= E) return;
    int s = (int)ei[wid];                  // src  (fits in i32: N=50000)
    int d = (int)ei[(long long)E + wid];   // dst
    float w = ew[wid];
    const float4* xs = (const float4*)(x + (size_t)s * C);
    float* ad = aggr + (size_t)d * C;
    for (int c4 = lane; c4 * 4 < C; c4 += 32) {
        float4 v = xs[c4];
        int c = c4 * 4;
        atomicAdd(ad + c + 0, w * v.x);
        atomicAdd(ad + c + 1, w * v.y);
        atomicAdd(ad + c + 2, w * v.z);
        atomicAdd(ad + c + 3, w * v.w);
    }
}

// ---------------------------------------------------------------------------
// GEMM + bias: out[N,M] = A[N,K] @ W[K,M] + b[M]
// Block = 256 threads = 8 waves, owning a 64-col stripe of W staged in LDS.
// LDS layout is K-pair interleaved: pair p, col c holds the 8-byte unit
//   ( W[2p][col0+c], W[2p+1][col0+c] )
// so each WMMA B fragment is ONE aligned ds_load_b64 straight into an even
// VGPR pair (ka = k0 + 2*half is always even). Wave computes 16x64 (4 tiles),
// reusing the A fragment across 4 x V_WMMA_F32_16X16X4_F32.
// ISA layouts (wave32):
//   A 16x4 f32 : lanes 0-15 -> M=lane, v0/v1 = K0/K1 ; lanes 16-31 -> K2/K3
//   B 4x16 f32 : lanes 0-15 -> N=lane, v0/v1 = K0/K1 ; lanes 16-31 -> K2/K3
//   D 16x16 f32: VGPR r -> (M=r+8*half, N=lane&15)
// ---------------------------------------------------------------------------
__global__ __launch_bounds__(256)
void gemm_bias_wmma(const float* __restrict__ A, const float* __restrict__ W,
                    const float* __restrict__ bias, float* __restrict__ out,
                    int N, int K, int M) {
    __shared__ v2f Wlds[(MAX_K / 2) * TILE_COLS];   // 64 KB (5x fits in 320KB WGP)

    int tid  = threadIdx.x;
    int col0 = blockIdx.x * TILE_COLS;

    // --- cooperative stage: interleave K-pairs, conflict-free ds_store_b64 --
    {
        int npairs = (K >> 1) * TILE_COLS;          // (K/2)*64 v2f units
        for (int i = tid; i < npairs; i += 256) {
            int p = i >> 6;                          // K pair index
            int c = i & 63;                          // column within stripe
            const float* wp = W + (size_t)(2 * p) * M + col0 + c;
            v2f u;
            u.x = wp[0];                             // W[2p  ][col0+c]
            u.y = wp[M];                             // W[2p+1][col0+c]
            Wlds[i] = u;
        }
    }
    __syncthreads();

    int wave = tid >> 5;
    int lane = tid & 31;
    int rowTile = blockIdx.y * 8 + wave;
    if (rowTile * 16 >= N) return;                   // wave-uniform exit

    int row0 = rowTile << 4;
    int half = lane >> 4;
    int lr   = lane & 15;

    v8f acc[4];
    #pragma unroll
    for (int t = 0; t < 4; ++t) acc[t] = (v8f){0.f,0.f,0.f,0.f,0.f,0.f,0.f,0.f};

    const float* Arow = A + (size_t)(row0 + lr) * K;

    for (int k0 = 0; k0 < K; k0 += 4) {
        int ka = k0 + 2 * half;                      // even
        v2f a;
        a.x = Arow[ka];
        a.y = Arow[ka + 1];
        const v2f* lb = Wlds + (ka >> 1) * TILE_COLS + lr;
        #pragma unroll
        for (int t = 0; t < 4; ++t) {
            v2f b = lb[t * 16];                      // one ds_load_b64
            acc[t] = __builtin_amdgcn_wmma_f32_16x16x4_f32(
                false, a, false, b, (short)0, acc[t], false, false);
        }
    }

    // --- epilogue: bias add + store 16x64 ----------------------------------
    #pragma unroll
    for (int t = 0; t < 4; ++t) {
        int cc = col0 + t * 16 + lr;
        float bv = bias[cc];
        #pragma unroll
        for (int r = 0; r < 8; ++r)
            out[(size_t)(row0 + r + 8 * half) * M + cc] = acc[t][r] + bv;
    }
}

// ---------------------------------------------------------------------------
// per-column sum / sum-of-squares (blockDim.x == M; each block: 256-row chunk)
// ---------------------------------------------------------------------------
__global__ void col_stats(const float* __restrict__ X, float* __restrict__ stats,
                          int N, int M, int rowsPerBlock) {
    int c = threadIdx.x;
    int r0 = blockIdx.x * rowsPerBlock;
    int r1 = r0 + rowsPerBlock;
    if (r1 > N) r1 = N;
    float s = 0.f, ss = 0.f;
    for (int r = r0; r < r1; ++r) {
        float v = X[(size_t)r * M + c];
        s += v;
        ss += v * v;
    }
    atomicAdd(&stats[c], s);
    atomicAdd(&stats[M + c], ss);
}

// ---------------------------------------------------------------------------
// y = relu((x - mean) * rsqrt(var + eps) * gamma + beta), biased variance
// ---------------------------------------------------------------------------
__global__ void bn_relu(const float* __restrict__ X, const float* __restrict__ stats,
                        const float* __restrict__ gamma, const float* __restrict__ beta,
                        float* __restrict__ Y, int N, int M) {
    long long i = (long long)blockIdx.x * blockDim.x + threadIdx.x;
    if (i >= (long long)N * M) return;
    int c = (int)(i % M);
    float invN = 1.0f / (float)N;
    float m = stats[c] * invN;
    float v = stats[M + c] * invN - m * m;
    float inv = rsqrtf(v + 1e-5f);
    float h = (X[i] - m) * inv * gamma[c] + beta[c];
    Y[i] = h > 0.f ? h : 0.f;
}

// ---------------------------------------------------------------------------
extern "C" void kernel_launch(void* const* d_in, const int* in_sizes, int n_in,
                              void* d_out, int out_size, void* d_ws, size_t ws_size,
                              hipStream_t stream) {
    (void)in_sizes; (void)n_in; (void)out_size; (void)ws_size;
    const int N = 50000, E = 800000, C = 128, H = 256;

    const float*     x  = (const float*)d_in[0];
    const long long* ei = (const long long*)d_in[1];
    const float*     ew = (const float*)d_in[2];
    const float* W1_0 = (const float*)d_in[3];
    const float* b1_0 = (const float*)d_in[4];
    const float* g1_0 = (const float*)d_in[5];
    const float* be1_0= (const float*)d_in[6];
    const float* W2_0 = (const float*)d_in[7];
    const float* b2_0 = (const float*)d_in[8];
    const float* g_0  = (const float*)d_in[9];
    const float* be_0 = (const float*)d_in[10];
    const float* W1_1 = (const float*)d_in[11];
    const float* b1_1 = (const float*)d_in[12];
    const float* g1_1 = (const float*)d_in[13];
    const float* be1_1= (const float*)d_in[14];
    const float* W2_1 = (const float*)d_in[15];
    const float* b2_1 = (const float*)d_in[16];
    const float* g_1  = (const float*)d_in[17];
    const float* be_1 = (const float*)d_in[18];

    float* bufB  = (float*)d_ws;                       // N*H floats
    float* bufC  = bufB + (size_t)N * H;               // N*H floats
    float* stats = bufC + (size_t)N * H;               // 2*H floats
    float* out   = (float*)d_out;

    auto zero = [&](float* p, long long n) {
        fill_zero<<<(unsigned)((n + 255) / 256), 256, 0, stream>>>(p, n);
    };
    auto scatter = [&](const float* xin, float* aggr, int c) {
        long long threads = (long long)E * 32;
        scatter_edges<<<(unsigned)(threads / 256), 256, 0, stream>>>(xin, ei, ew, aggr, E, c);
    };
    auto gemm = [&](const float* A, const float* W, const float* b, float* O,
                    int n, int k, int m) {
        dim3 grid(m / TILE_COLS, (n / 16 + 7) / 8);    // stripes x rowBlocks
        gemm_bias_wmma<<<grid, 256, 0, stream>>>(A, W, b, O, n, k, m);
    };
    auto bn = [&](const float* X, const float* g, const float* be, float* Y, int m) {
        zero(stats, 2 * m);
        col_stats<<<(N + 255) / 256, m, 0, stream>>>(X, stats, N, m, 256);
        long long tot = (long long)N * m;
        bn_relu<<<(unsigned)((tot + 255) / 256), 256, 0, stream>>>(X, stats, g, be, Y, N, m);
    };

    // ---------------- layer 0: 128 -> 256 ----------------
    zero(bufB, (long long)N * C);
    scatter(x, bufB, C);                       // aggr0 [N,128] in bufB
    gemm(bufB, W1_0, b1_0, bufC, N, C, H);     // T1 [N,256] in bufC
    bn(bufC, g1_0, be1_0, bufC, H);            // in-place BN+ReLU
    gemm(bufC, W2_0, b2_0, bufB, N, H, H);     // T2 [N,256] in bufB
    bn(bufB, g_0, be_0, bufB, H);              // h0 [N,256] in bufB

    // ---------------- layer 1: 256 -> 128 ----------------
    zero(bufC, (long long)N * H);
    scatter(bufB, bufC, H);                    // aggr1 [N,256] in bufC
    gemm(bufC, W1_1, b1_1, bufB, N, H, C);     // T1' [N,128] in bufB
    bn(bufB, g1_1, be1_1, bufB, C);            // in-place BN+ReLU
    gemm(bufB, W2_1, b2_1, bufC, N, C, C);     // T2' [N,128] in bufC
    bn(bufC, g_1, be_1, out, C);               // final BN+ReLU -> d_out
}